// VanillaGCN_61967788146722
// MI455X (gfx1250) — compile-verified
//
#include <hip/hip_runtime.h>

#define CHN 256
#define EPS 1e-5f

typedef float v2f __attribute__((ext_vector_type(2)));
typedef float v8f __attribute__((ext_vector_type(8)));

// ---------------------------------------------------------------- utilities
__global__ __launch_bounds__(256) void zero_kernel(float* __restrict__ p, int n4) {
    const int i = blockIdx.x * 256 + threadIdx.x;
    if (i < n4) reinterpret_cast<float4*>(p)[i] = make_float4(0.f, 0.f, 0.f, 0.f);
}

__global__ __launch_bounds__(256) void deg_kernel(const int* __restrict__ src,
                                                  const int* __restrict__ dst,
                                                  float* __restrict__ deg_out,
                                                  float* __restrict__ deg_in, int E) {
    const int e = blockIdx.x * 256 + threadIdx.x;
    if (e < E) {
        atomicAdd(deg_out + src[e], 1.0f);
        atomicAdd(deg_in + dst[e], 1.0f);
    }
}

// in-place: deg -> (deg>0 ? rsqrt(deg) : 0)
__global__ __launch_bounds__(256) void norm_kernel(float* __restrict__ a,
                                                   float* __restrict__ b, int n) {
    const int i = blockIdx.x * 256 + threadIdx.x;
    if (i < n) {
        const float da = a[i];
        const float db = b[i];
        a[i] = da > 0.f ? rsqrtf(da) : 0.f;
        b[i] = db > 0.f ? rsqrtf(db) : 0.f;
    }
}

// out[row,c] = x[row,c] * ns[row]
__global__ __launch_bounds__(256) void scale_rows_kernel(const float* __restrict__ x,
                                                         const float* __restrict__ ns,
                                                         float* __restrict__ out, int n4) {
    const int i = blockIdx.x * 256 + threadIdx.x;
    if (i >= n4) return;
    const int row = (i << 2) >> 8;   // CHN == 256
    const float s = ns[row];
    float4 v = reinterpret_cast<const float4*>(x)[i];
    v.x *= s; v.y *= s; v.z *= s; v.w *= s;
    reinterpret_cast<float4*>(out)[i] = v;
}

// ------------------------------------------------------- edge scatter (L2-resident)
// one wave per edge: gather 256 floats of src row, atomic-add into dst row
__global__ __launch_bounds__(256) void scatter_kernel(const float* __restrict__ hs,
                                                      const int* __restrict__ src,
                                                      const int* __restrict__ dst,
                                                      float* __restrict__ agg, int E) {
    const int e = blockIdx.x * 8 + (threadIdx.x >> 5);
    if (e >= E) return;
    const int lane = threadIdx.x & 31;
    const int s = src[e];
    const int d = dst[e];
    const float4* in = reinterpret_cast<const float4*>(hs + (size_t)s * CHN);
    float* out = agg + (size_t)d * CHN;
#pragma unroll
    for (int j = 0; j < 2; ++j) {
        const int q = lane + 32 * j;          // float4 index within the row
        const float4 v = in[q];
        atomicAdd(out + q * 4 + 0, v.x);
        atomicAdd(out + q * 4 + 1, v.y);
        atomicAdd(out + q * 4 + 2, v.z);
        atomicAdd(out + q * 4 + 3, v.w);
    }
}

// ---------------------------------------------------------------- WMMA GEMM
// C[M,NCOLS] = (rowscale ? diag(rowscale) : I) * A[M,256] @ W[256,NCOLS] + bias
// Block: 64 rows x all NCOLS columns, NCOLS*2 threads (NCOLS/16 waves).
// Each wave: one 16-col tile, 4 row tiles -> 4 f32 16x16x4 WMMA accumulators.
// In-place safe (Cout may alias Ain): A is read only during LDS staging, and a
// block touches only its own 64 rows.
template <int NCOLS, bool STATS>
__global__ __launch_bounds__(NCOLS * 2) void gemm_rows64_kernel(
    const float* __restrict__ Ain,        // [M,256]
    const float* __restrict__ rowscale,   // [M] or nullptr
    const float* __restrict__ Wmat,       // [256,NCOLS]
    const float* __restrict__ bias,       // [NCOLS]
    float* __restrict__ Cout,             // [M,NCOLS]
    float* __restrict__ stats,            // [2*NCOLS]: col sums, col sumsqs
    int M) {
    constexpr int K = CHN;
    constexpr int LDSS = K + 2;           // 258: even (b64-aligned), conflict-free
    constexpr int NTHREADS = NCOLS * 2;
    __shared__ float Ap[64 * LDSS];

    const int tid  = threadIdx.x;
    const int wave = tid >> 5;
    const int lane = tid & 31;
    const int half = lane >> 4;           // K-half / M-half selector
    const int l16  = lane & 15;
    const int row0 = blockIdx.x * 64;

    // Stage 64x256 A panel into LDS, applying rowscale; zero-fill OOB rows.
    for (int i = tid; i < 64 * (K / 4); i += NTHREADS) {
        const int r  = i >> 6;
        const int kc = (i & 63) << 2;
        const int row = row0 + r;
        float4 v = make_float4(0.f, 0.f, 0.f, 0.f);
        float s = 0.f;
        if (row < M) {
            v = reinterpret_cast<const float4*>(Ain + (size_t)row * K)[kc >> 2];
            s = rowscale ? rowscale[row] : 1.0f;
        }
        float* lp = &Ap[r * LDSS + kc];
        lp[0] = v.x * s; lp[1] = v.y * s; lp[2] = v.z * s; lp[3] = v.w * s;
    }
    __syncthreads();

    const int colbase = wave * 16;
    const float* wp = Wmat + colbase + l16;

    v8f acc[4] = {};
    for (int k = 0; k < K; k += 4) {
        const int ka = k + half * 2;
        // B fragment (4x16 f32): VGPR v <-> K = ka+v, lane-half splits K range
        v2f b;
        b.x = wp[(size_t)ka * NCOLS];
        b.y = wp[(size_t)(ka + 1) * NCOLS];
#pragma unroll
        for (int r = 0; r < 4; ++r) {
            // A fragment (16x4 f32): row = l16, K = ka..ka+1 per lane-half
            const v2f a = *reinterpret_cast<const v2f*>(&Ap[(r * 16 + l16) * LDSS + ka]);
            acc[r] = __builtin_amdgcn_wmma_f32_16x16x4_f32(
                false, a, false, b, (short)0, acc[r], false, false);
        }
    }

    // Epilogue: bias add, store, fused per-column BN statistics.
    const float bcol = bias[colbase + l16];
    float lsum = 0.f, lsq = 0.f;
#pragma unroll
    for (int r = 0; r < 4; ++r) {
#pragma unroll
        for (int v = 0; v < 8; ++v) {
            const int row = row0 + r * 16 + v + 8 * half;  // C layout: M = v (+8 for hi half)
            if (row < M) {
                const float val = acc[r][v] + bcol;
                Cout[(size_t)row * NCOLS + colbase + l16] = val;
                if (STATS) { lsum += val; lsq += val * val; }
            }
        }
    }
    if (STATS) {
        atomicAdd(stats + colbase + l16, lsum);
        atomicAdd(stats + NCOLS + colbase + l16, lsq);
    }
}

// ---------------------------------------------------------------- batch norm
__global__ void bn_finalize_kernel(const float* __restrict__ stats,
                                   const float* __restrict__ g,
                                   const float* __restrict__ beta,
                                   float* __restrict__ ab, float invN) {
    const int c = threadIdx.x;            // 256 threads
    const float mean = stats[c] * invN;
    const float var  = fmaxf(stats[CHN + c] * invN - mean * mean, 0.f);
    const float a = g[c] * rsqrtf(var + EPS);
    ab[c]       = a;
    ab[CHN + c] = beta[c] - mean * a;
}

// out = relu(h*a + b) * (rowscale ? rowscale[row] : 1)
__global__ __launch_bounds__(256) void bn_apply_kernel(const float* __restrict__ hlin,
                                                       const float* __restrict__ ab,
                                                       const float* __restrict__ rowscale,
                                                       float* __restrict__ out, int n4) {
    const int i = blockIdx.x * 256 + threadIdx.x;
    if (i >= n4) return;
    const int base = i << 2;
    const int row = base >> 8;
    const int c = base & (CHN - 1);
    const float s = rowscale ? rowscale[row] : 1.0f;
    const float4 v = reinterpret_cast<const float4*>(hlin)[i];
    float4 o;
    o.x = fmaxf(fmaf(v.x, ab[c + 0], ab[CHN + c + 0]), 0.f) * s;
    o.y = fmaxf(fmaf(v.y, ab[c + 1], ab[CHN + c + 1]), 0.f) * s;
    o.z = fmaxf(fmaf(v.z, ab[c + 2], ab[CHN + c + 2]), 0.f) * s;
    o.w = fmaxf(fmaf(v.w, ab[c + 3], ab[CHN + c + 3]), 0.f) * s;
    reinterpret_cast<float4*>(out)[i] = o;
}

// ---------------------------------------------------------------- launch
extern "C" void kernel_launch(void* const* d_in, const int* in_sizes, int n_in,
                              void* d_out, int out_size, void* d_ws, size_t ws_size,
                              hipStream_t stream) {
    (void)n_in; (void)out_size; (void)ws_size;
    const float* x    = (const float*)d_in[0];
    const int*   src  = (const int*)d_in[1];
    const int*   dst  = (const int*)d_in[2];
    const float* W[3]    = {(const float*)d_in[3],  (const float*)d_in[7],  (const float*)d_in[11]};
    const float* bvec[3] = {(const float*)d_in[4],  (const float*)d_in[8],  (const float*)d_in[12]};
    const float* g[3]    = {(const float*)d_in[5],  (const float*)d_in[9],  (const float*)d_in[13]};
    const float* beta[3] = {(const float*)d_in[6],  (const float*)d_in[10], (const float*)d_in[14]};
    const float* fcW = (const float*)d_in[15];
    const float* fcb = (const float*)d_in[16];

    const int N = in_sizes[0] / CHN;
    const int E = in_sizes[1];

    // Workspace layout (fp32): norms | bufA (h_scaled/h) | bufB (agg & in-place h_lin)
    float* ws = (float*)d_ws;
    float* norm_src = ws;                 ws += N;
    float* norm_dst = ws;                 ws += N;
    float* bufA     = ws;                 ws += (size_t)N * CHN;
    float* bufB     = ws;                 ws += (size_t)N * CHN;
    float* stats    = ws;                 ws += 2 * CHN;
    float* ab       = ws;                 ws += 2 * CHN;

    const int n4feat   = N * CHN / 4;
    const int featBlks = (n4feat + 255) / 256;
    const int rowBlks  = (N + 63) / 64;
    const int edgeBlks = (E + 7) / 8;

    // Degree norms
    zero_kernel<<<(2 * N / 4 + 255) / 256, 256, 0, stream>>>(norm_src, 2 * N / 4);
    deg_kernel<<<(E + 255) / 256, 256, 0, stream>>>(src, dst, norm_src, norm_dst, E);
    norm_kernel<<<(N + 255) / 256, 256, 0, stream>>>(norm_src, norm_dst, N);

    // h_scaled = x * norm_src
    scale_rows_kernel<<<featBlks, 256, 0, stream>>>(x, norm_src, bufA, n4feat);

    for (int l = 0; l < 3; ++l) {
        // agg = scatter_add(h_scaled[src] -> dst)
        zero_kernel<<<featBlks, 256, 0, stream>>>(bufB, n4feat);
        scatter_kernel<<<edgeBlks, 256, 0, stream>>>(bufA, src, dst, bufB, E);
        // h_lin = (agg * norm_dst) @ W + b   (in place over bufB), fused BN stats
        zero_kernel<<<1, 256, 0, stream>>>(stats, 2 * CHN / 4);
        gemm_rows64_kernel<CHN, true><<<rowBlks, CHN * 2, 0, stream>>>(
            bufB, norm_dst, W[l], bvec[l], bufB, stats, N);
        bn_finalize_kernel<<<1, CHN, 0, stream>>>(stats, g[l], beta[l], ab, 1.0f / (float)N);
        // h_scaled(next) = relu(bn(h_lin)) * norm_src   (last layer: unscaled h)
        bn_apply_kernel<<<featBlks, 256, 0, stream>>>(
            bufB, ab, (l < 2) ? norm_src : nullptr, bufA, n4feat);
    }

    // out = h @ fcW + fcb
    gemm_rows64_kernel<64, false><<<rowBlks, 64 * 2, 0, stream>>>(
        bufA, nullptr, fcW, fcb, (float*)d_out, nullptr, N);
}